// GenericMAE_4956392259938
// MI455X (gfx1250) — compile-verified
//
#include <hip/hip_runtime.h>

typedef __attribute__((ext_vector_type(2))) float v2f;
typedef __attribute__((ext_vector_type(4))) float v4f;
typedef __attribute__((ext_vector_type(8))) float v8f;

#define HDIM 1024       // hidden size H == N
#define PDIM 768        // patch dim PD
#define NPATCH 1024     // N
#define NMASK 768       // M
#define NUNMASK 256     // N - M
#define NBATCH 32
#define LN_EPS 1e-5f

// ---------------------------------------------------------------------------
// GEMM1: E[256, H] = pos_emb[masked_idx[m], :] @ W_enc[H,H] + b_enc
// One wave computes a 16x64 tile: 4 WMMA accumulators, A fragment reused.
// ---------------------------------------------------------------------------
__global__ __launch_bounds__(32)
void enc_gemm_kernel(const float* __restrict__ pos_emb,
                     const float* __restrict__ W_enc,
                     const float* __restrict__ b_enc,
                     const int*   __restrict__ masked_idx,
                     float*       __restrict__ E) {
    const int lane  = threadIdx.x;
    const int half  = lane >> 4;      // 0 or 1
    const int l15   = lane & 15;
    const int kboff = half << 1;      // K offset inside a K=4 step (0 or 2)
    const int tile_m = blockIdx.y * 16;   // 0..240
    const int tile_n = blockIdx.x * 64;   // 0..960

    const int arow = masked_idx[tile_m + l15];
    const float* __restrict__ aptr = pos_emb + (size_t)arow * HDIM;

    v8f acc0 = {}, acc1 = {}, acc2 = {}, acc3 = {};

    for (int kb = 0; kb < HDIM; kb += 4) {
        // A fragment 16x4 f32: lane (m=l15) holds {A[m,kb+kboff], A[m,kb+kboff+1]}
        v2f a;
        a.x = aptr[kb + kboff];
        a.y = aptr[kb + kboff + 1];

        const float* __restrict__ bbase =
            W_enc + (size_t)(kb + kboff) * HDIM + tile_n + l15;
        v2f b0, b1, b2, b3;
        b0.x = bbase[0];   b0.y = bbase[HDIM + 0];
        b1.x = bbase[16];  b1.y = bbase[HDIM + 16];
        b2.x = bbase[32];  b2.y = bbase[HDIM + 32];
        b3.x = bbase[48];  b3.y = bbase[HDIM + 48];

        acc0 = __builtin_amdgcn_wmma_f32_16x16x4_f32(false, a, false, b0, (short)0, acc0, false, false);
        acc1 = __builtin_amdgcn_wmma_f32_16x16x4_f32(false, a, false, b1, (short)0, acc1, false, false);
        acc2 = __builtin_amdgcn_wmma_f32_16x16x4_f32(false, a, false, b2, (short)0, acc2, false, false);
        acc3 = __builtin_amdgcn_wmma_f32_16x16x4_f32(false, a, false, b3, (short)0, acc3, false, false);
    }

    // D layout: vgpr i, lanes 0-15 -> M = i, lanes 16-31 -> M = i+8; N = l15
    const int row0 = tile_m + half * 8;
    #pragma unroll
    for (int i = 0; i < 8; ++i) {
        const int row = row0 + i;
        float* __restrict__ erow = E + (size_t)row * HDIM + tile_n + l15;
        erow[0]  = acc0[i] + b_enc[tile_n + l15 + 0];
        erow[16] = acc1[i] + b_enc[tile_n + l15 + 16];
        erow[32] = acc2[i] + b_enc[tile_n + l15 + 32];
        erow[48] = acc3[i] + b_enc[tile_n + l15 + 48];
    }
}

// ---------------------------------------------------------------------------
// LayerNorm in-place on E rows (256 rows x 1024 cols). One 256-thread block
// (8 waves) per row; wave32 shuffle reduce + LDS cross-wave reduce.
// ---------------------------------------------------------------------------
__global__ __launch_bounds__(256)
void layernorm_kernel(float* __restrict__ E,
                      const float* __restrict__ gamma,
                      const float* __restrict__ beta) {
    __shared__ float s_sum[8];
    __shared__ float s_sq[8];

    float* __restrict__ p = E + (size_t)blockIdx.x * HDIM;

    float x[4];
    float sum = 0.f, sq = 0.f;
    #pragma unroll
    for (int i = 0; i < 4; ++i) {
        x[i] = p[threadIdx.x + i * 256];
        sum += x[i];
        sq  += x[i] * x[i];
    }
    // wave32 reduction
    #pragma unroll
    for (int o = 16; o > 0; o >>= 1) {
        sum += __shfl_xor(sum, o, 32);
        sq  += __shfl_xor(sq,  o, 32);
    }
    const int wid = threadIdx.x >> 5;
    if ((threadIdx.x & 31) == 0) { s_sum[wid] = sum; s_sq[wid] = sq; }
    __syncthreads();

    float t1 = 0.f, t2 = 0.f;
    #pragma unroll
    for (int i = 0; i < 8; ++i) { t1 += s_sum[i]; t2 += s_sq[i]; }

    const float mu  = t1 * (1.0f / HDIM);
    const float var = t2 * (1.0f / HDIM) - mu * mu;
    const float inv = rsqrtf(var + LN_EPS);

    #pragma unroll
    for (int i = 0; i < 4; ++i) {
        const int c = threadIdx.x + i * 256;
        p[c] = (x[i] - mu) * inv * gamma[c] + beta[c];
    }
}

// ---------------------------------------------------------------------------
// Build recon[1024, H]: masked rows copy pos_emb, unmasked rows copy normed.
// The index arrays form a permutation -> every row is written exactly once.
// ---------------------------------------------------------------------------
__global__ __launch_bounds__(256)
void build_recon_kernel(const float* __restrict__ pos_emb,
                        const float* __restrict__ normed,
                        const int*   __restrict__ masked_idx,
                        const int*   __restrict__ unmasked_idx,
                        float*       __restrict__ recon) {
    const int blk = blockIdx.x;   // 0..1023
    const float* __restrict__ src;
    int r;
    if (blk < NMASK) {
        r = masked_idx[blk];
        src = pos_emb + (size_t)r * HDIM;
    } else {
        const int j = blk - NMASK;     // 0..255
        r = unmasked_idx[j];
        src = normed + (size_t)j * HDIM;
    }
    const v4f* __restrict__ s4 = (const v4f*)src;
    v4f* __restrict__ d4 = (v4f*)(recon + (size_t)r * HDIM);
    d4[threadIdx.x] = s4[threadIdx.x];   // 256 threads * 16B = 4KB row
}

// ---------------------------------------------------------------------------
// GEMM2: Omat[d, s] = sum_c W_dec[c, d] * recon[c, s] + b_dec[d]
// A = W_dec^T (768 x 1024, strided column loads), B = recon.
// One wave per 16x64 tile.
// ---------------------------------------------------------------------------
__global__ __launch_bounds__(32)
void dec_gemm_kernel(const float* __restrict__ W_dec,
                     const float* __restrict__ recon,
                     const float* __restrict__ b_dec,
                     float*       __restrict__ Omat) {
    const int lane  = threadIdx.x;
    const int half  = lane >> 4;
    const int l15   = lane & 15;
    const int kboff = half << 1;
    const int tile_m = blockIdx.y * 16;   // d tiles: 0..752
    const int tile_n = blockIdx.x * 64;   // s tiles: 0..960

    const int d = tile_m + l15;

    v8f acc0 = {}, acc1 = {}, acc2 = {}, acc3 = {};

    for (int kb = 0; kb < NPATCH; kb += 4) {
        // A[d, c] = W_dec[c*PD + d]  (transposed access, stride PD)
        v2f a;
        a.x = W_dec[(size_t)(kb + kboff)     * PDIM + d];
        a.y = W_dec[(size_t)(kb + kboff + 1) * PDIM + d];

        const float* __restrict__ bbase =
            recon + (size_t)(kb + kboff) * HDIM + tile_n + l15;
        v2f b0, b1, b2, b3;
        b0.x = bbase[0];   b0.y = bbase[HDIM + 0];
        b1.x = bbase[16];  b1.y = bbase[HDIM + 16];
        b2.x = bbase[32];  b2.y = bbase[HDIM + 32];
        b3.x = bbase[48];  b3.y = bbase[HDIM + 48];

        acc0 = __builtin_amdgcn_wmma_f32_16x16x4_f32(false, a, false, b0, (short)0, acc0, false, false);
        acc1 = __builtin_amdgcn_wmma_f32_16x16x4_f32(false, a, false, b1, (short)0, acc1, false, false);
        acc2 = __builtin_amdgcn_wmma_f32_16x16x4_f32(false, a, false, b2, (short)0, acc2, false, false);
        acc3 = __builtin_amdgcn_wmma_f32_16x16x4_f32(false, a, false, b3, (short)0, acc3, false, false);
    }

    const int row0 = tile_m + half * 8;
    #pragma unroll
    for (int i = 0; i < 8; ++i) {
        const int row = row0 + i;            // d index -> bias along rows
        const float bias = b_dec[row];
        float* __restrict__ orow = Omat + (size_t)row * NPATCH + tile_n + l15;
        orow[0]  = acc0[i] + bias;
        orow[16] = acc1[i] + bias;
        orow[32] = acc2[i] + bias;
        orow[48] = acc3[i] + bias;
    }
}

// ---------------------------------------------------------------------------
// Broadcast Omat (768x1024 = 3MB) to all 32 batches of d_out (96MB stores).
// Memory-bound stage: float4 stores, source stays resident in L2.
// ---------------------------------------------------------------------------
__global__ __launch_bounds__(256)
void broadcast_kernel(const float* __restrict__ Omat,
                      float*       __restrict__ out) {
    const size_t i = (size_t)blockIdx.x * 256 + threadIdx.x;   // float4 index
    const v4f v = ((const v4f*)Omat)[i];
    ((v4f*)out)[(size_t)blockIdx.y * (PDIM * NPATCH / 4) + i] = v;
}

// ---------------------------------------------------------------------------
extern "C" void kernel_launch(void* const* d_in, const int* in_sizes, int n_in,
                              void* d_out, int out_size, void* d_ws, size_t ws_size,
                              hipStream_t stream) {
    // setup_inputs order:
    // 0:x 1:W_patch 2:b_patch 3:pos_emb 4:W_enc 5:b_enc 6:gamma 7:beta
    // 8:W_dec 9:b_dec 10:masked_idx 11:unmasked_idx
    const float* pos_emb = (const float*)d_in[3];
    const float* W_enc   = (const float*)d_in[4];
    const float* b_enc   = (const float*)d_in[5];
    const float* gamma   = (const float*)d_in[6];
    const float* beta    = (const float*)d_in[7];
    const float* W_dec   = (const float*)d_in[8];
    const float* b_dec   = (const float*)d_in[9];
    const int*   midx    = (const int*)d_in[10];
    const int*   uidx    = (const int*)d_in[11];

    float* E     = (float*)d_ws;                         // 256*1024 fp32 (1 MB)
    float* recon = E + (size_t)NUNMASK * HDIM;           // 1024*1024 fp32 (4 MB)
    float* Omat  = recon + (size_t)NPATCH * HDIM;        // 768*1024 fp32 (3 MB)
    float* out   = (float*)d_out;

    // 1) encoder GEMM on the 256 rows that survive to the output
    enc_gemm_kernel<<<dim3(HDIM / 64, NUNMASK / 16), 32, 0, stream>>>(
        pos_emb, W_enc, b_enc, midx, E);

    // 2) layernorm (in place)
    layernorm_kernel<<<NUNMASK, 256, 0, stream>>>(E, gamma, beta);

    // 3) scatter permutation into recon
    build_recon_kernel<<<NPATCH, 256, 0, stream>>>(pos_emb, E, midx, uidx, recon);

    // 4) decoder GEMM: Omat = W_dec^T @ recon + b_dec
    dec_gemm_kernel<<<dim3(NPATCH / 64, PDIM / 16), 32, 0, stream>>>(
        W_dec, recon, b_dec, Omat);

    // 5) batch-broadcast (output is batch-invariant)
    broadcast_kernel<<<dim3(PDIM * NPATCH / 4 / 256, NBATCH), 256, 0, stream>>>(
        Omat, out);
}